// Temporal_Feature_Aggregation_46651934769849
// MI455X (gfx1250) — compile-verified
//
#include <hip/hip_runtime.h>
#include <hip/hip_bf16.h>

// ---------------------------------------------------------------------------
// Types for WMMA fragments (CDNA5 / gfx1250, wave32)
// ---------------------------------------------------------------------------
typedef __attribute__((ext_vector_type(2))) float v2f;
typedef __attribute__((ext_vector_type(8))) float v8f;

#define BK   32   // K-depth of an LDS stage
#define BKP  36   // padded LDS row stride (floats): 36*i mod 64 distinct for i<16, float4-aligned

// ---------------------------------------------------------------------------
// CDNA5 async Global->LDS copy (bypasses VGPRs, tracked by ASYNCcnt).
// lds_dst low 32 bits == LDS byte offset.
// ---------------------------------------------------------------------------
__device__ __forceinline__ void async_copy_b128(const float* gsrc, float* lds_dst) {
    unsigned ldsOff = (unsigned)(size_t)lds_dst;
    asm volatile("global_load_async_to_lds_b128 %0, %1, off"
                 :
                 : "v"(ldsOff), "v"(gsrc)
                 : "memory");
}

__device__ __forceinline__ void wait_async_zero() {
    asm volatile("s_wait_asynccnt 0x0" ::: "memory");
}

// ---------------------------------------------------------------------------
// Generic split-K GEMM:  C[48 x N] += A[48 x Ktot] * W[N x Ktot]^T
// over this block's K-chunk [blockIdx.y*KC, +KC).
// Block = 96 threads = 3 waves; wave w owns m-tile w (rows 16w..16w+15).
// blockIdx.x = n-tile (16 columns). Split-K partials merged with f32 atomics.
// Tiles staged via GLOBAL_LOAD_ASYNC_TO_LDS_B128; fragments via ds_load.
// ---------------------------------------------------------------------------
__global__ __launch_bounds__(96)
void gemm48_nt_wmma(const float* __restrict__ A, const float* __restrict__ W,
                    float* __restrict__ C, int N, int Ktot, int KC) {
    __shared__ float lA[48 * BKP];
    __shared__ float lW[16 * BKP];

    const int tid   = threadIdx.x;
    const int lane  = tid & 31;
    const int wave  = tid >> 5;           // m-tile index 0..2
    const int ntile = blockIdx.x;
    const int kbase = blockIdx.y * KC;

    const int lan16 = lane & 15;
    const int khalf = (lane >> 4) * 2;    // lanes 0-15 -> K {0,1}, lanes 16-31 -> K {2,3}

    v8f acc = {};

    for (int kk = 0; kk < KC; kk += BK) {
        const int kg = kbase + kk;

        // Async staging: A tile 48 x BK (float4 per lane, coalesced per row-chunk)
        for (int idx = tid; idx < 48 * (BK / 4); idx += 96) {
            const int r  = idx / (BK / 4);
            const int c4 = (idx % (BK / 4)) * 4;
            async_copy_b128(A + (size_t)r * Ktot + kg + c4, lA + r * BKP + c4);
        }
        // Async staging: W tile 16 x BK (rows = output columns of this n-tile)
        for (int idx = tid; idx < 16 * (BK / 4); idx += 96) {
            const int r  = idx / (BK / 4);
            const int c4 = (idx % (BK / 4)) * 4;
            async_copy_b128(W + (size_t)(ntile * 16 + r) * Ktot + kg + c4,
                            lW + r * BKP + c4);
        }
        wait_async_zero();
        __syncthreads();

        const float* aRow = lA + (wave * 16 + lan16) * BKP + khalf;
        const float* bRow = lW + lan16 * BKP + khalf;
#pragma unroll
        for (int k4 = 0; k4 < BK; k4 += 4) {
            v2f a = *(const v2f*)(aRow + k4);   // A frag: M=lane%16, K = k4 + khalf + {0,1}
            v2f b = *(const v2f*)(bRow + k4);   // B frag: N=lane%16, K = k4 + khalf + {0,1}
            acc = __builtin_amdgcn_wmma_f32_16x16x4_f32(
                false, a, false, b, (short)0, acc, false, false);
        }
        __syncthreads();
    }

    // D layout: VGPR r -> M = r + 8*(lane>=16), N = lane%16
    const int n     = ntile * 16 + lan16;
    const int mbase = wave * 16 + (lane >> 4) * 8;
#pragma unroll
    for (int r = 0; r < 8; ++r) {
        atomicAdd(C + (size_t)(mbase + r) * N + n, acc[r]);
    }
}

// ---------------------------------------------------------------------------
// Hf epilogue: Hf = relu(Hf_acc + b_T3)   (48 x N)
// ---------------------------------------------------------------------------
__global__ void bias_relu48(const float* __restrict__ acc,
                            const float* __restrict__ bias,
                            float* __restrict__ Y, int N) {
    const int i = blockIdx.x * blockDim.x + threadIdx.x;
    if (i >= 48 * N) return;
    Y[i] = fmaxf(acc[i] + bias[i % N], 0.0f);
}

// ---------------------------------------------------------------------------
// Tiny input projections: Y[48 x N] = relu(X[48 x K] @ Wt[N x K]^T + b)
// ---------------------------------------------------------------------------
__global__ void proj_relu(const float* __restrict__ X, const float* __restrict__ Wt,
                          const float* __restrict__ bias, float* __restrict__ Y,
                          int K, int N) {
    const int i = blockIdx.x * blockDim.x + threadIdx.x;
    if (i >= 48 * N) return;
    const int r = i / N, o = i % N;
    float s = bias[o];
    const float* xr = X + (size_t)r * K;
    const float* wr = Wt + (size_t)o * K;
    for (int k = 0; k < K; ++k) s = fmaf(xr[k], wr[k], s);
    Y[i] = fmaxf(s, 0.0f);
}

// ---------------------------------------------------------------------------
// Gram matrices [B,3,3] (+softmax for the first three) -> stacked S[80][9]
// blockIdx.x = which*16 + b ; which: 0 Hf, 1 bpf, 2 wf, 3 (Q1,Q2), 4 (P3,P4)
// ---------------------------------------------------------------------------
__global__ __launch_bounds__(256)
void gram_softmax(const float* __restrict__ Hf, const float* __restrict__ bpf,
                  const float* __restrict__ wf,
                  const float* __restrict__ Q1, const float* __restrict__ Q2,
                  const float* __restrict__ P3, const float* __restrict__ P4,
                  float* __restrict__ S) {
    const int b     = blockIdx.x & 15;
    const int which = blockIdx.x >> 4;

    const float *X, *Y;
    int K;
    switch (which) {
        case 0:  X = Hf;  Y = Hf;  K = 2048; break;
        case 1:  X = bpf; Y = bpf; K = 512;  break;
        case 2:  X = wf;  Y = wf;  K = 512;  break;
        case 3:  X = Q1;  Y = Q2;  K = 512;  break;
        default: X = P3;  Y = P4;  K = 1024; break;
    }

    __shared__ float red[256];
    __shared__ float m9[9];

    for (int p = 0; p < 9; ++p) {
        const int i = p / 3, j = p % 3;
        const float* xr = X + (size_t)(b * 3 + i) * K;
        const float* yr = Y + (size_t)(b * 3 + j) * K;
        float s = 0.0f;
        for (int k = threadIdx.x; k < K; k += 256) s = fmaf(xr[k], yr[k], s);
        red[threadIdx.x] = s;
        __syncthreads();
        for (int w = 128; w > 0; w >>= 1) {
            if (threadIdx.x < w) red[threadIdx.x] += red[threadIdx.x + w];
            __syncthreads();
        }
        if (threadIdx.x == 0) m9[p] = red[0];
        __syncthreads();
    }

    if (threadIdx.x == 0) {
        float out9[9];
        if (which < 3) {
            for (int i = 0; i < 3; ++i) {
                const float a0 = m9[i * 3], a1 = m9[i * 3 + 1], a2 = m9[i * 3 + 2];
                const float mx = fmaxf(a0, fmaxf(a1, a2));
                const float e0 = __expf(a0 - mx), e1 = __expf(a1 - mx), e2 = __expf(a2 - mx);
                const float inv = 1.0f / (e0 + e1 + e2);
                out9[i * 3 + 0] = e0 * inv;
                out9[i * 3 + 1] = e1 * inv;
                out9[i * 3 + 2] = e2 * inv;
            }
        } else {
            for (int p = 0; p < 9; ++p) out9[p] = m9[p];
        }
        for (int p = 0; p < 9; ++p) S[(which * 16 + b) * 9 + p] = out9[p];
    }
}

// ---------------------------------------------------------------------------
// phi6 (faithful batch-mixing reshape: row 5b+c of the stacked S[80][9])
// then mul[b,t,:] = phi6[b,t,:] @ P5[b]   (3x3 @ 3x1024)
// ---------------------------------------------------------------------------
__global__ __launch_bounds__(256)
void phi6_mul(const float* __restrict__ S, const float* __restrict__ Wphi6,
              const float* __restrict__ P5, float* __restrict__ mul) {
    const int b = blockIdx.x;
    __shared__ float ph[9];
    if (threadIdx.x < 9) {
        float s = 0.0f;
        for (int c = 0; c < 5; ++c) s = fmaf(Wphi6[c], S[(5 * b + c) * 9 + threadIdx.x], s);
        ph[threadIdx.x] = s;
    }
    __syncthreads();
    for (int k = threadIdx.x; k < 1024; k += 256) {
#pragma unroll
        for (int t = 0; t < 3; ++t) {
            float s = 0.0f;
#pragma unroll
            for (int i = 0; i < 3; ++i)
                s = fmaf(ph[t * 3 + i], P5[(size_t)(b * 3 + i) * 1024 + k], s);
            mul[(size_t)(b * 3 + t) * 1024 + k] = s;
        }
    }
}

// ---------------------------------------------------------------------------
// Final: Z[b,o,t] = Hf[b*3+t, o] + Racc[b*3+t, o]; written twice (outputs 0,2)
// ---------------------------------------------------------------------------
__global__ void final_z(const float* __restrict__ Hf, const float* __restrict__ Racc,
                        float* __restrict__ out) {
    const int i = blockIdx.x * blockDim.x + threadIdx.x;
    if (i >= 48 * 2048) return;
    const int r = i / 2048, o = i % 2048;
    const int b = r / 3, t = r % 3;
    const float z = Hf[i] + Racc[i];
    const size_t oidx = (size_t)b * 6144 + (size_t)o * 3 + t;
    out[oidx]               = z;
    out[98304 + 48 + oidx]  = z;
}

// Deterministic placeholder for the untied PRNG output (16x3 normals).
__global__ void fill_rand(float* __restrict__ out) {
    const int i = threadIdx.x;
    if (i >= 48) return;
    unsigned h = (unsigned)i * 2654435761u ^ 0x9E3779B9u;
    h ^= h >> 16; h *= 0x85EBCA6Bu; h ^= h >> 13;
    const float u1 = (float)((h & 0xFFFFFFu) + 1u) / 16777217.0f;
    unsigned h2 = h * 0xC2B2AE35u; h2 ^= h2 >> 16;
    const float u2 = (float)(h2 & 0xFFFFFFu) / 16777216.0f;
    out[98304 + i] = sqrtf(-2.0f * __logf(u1)) * __cosf(6.2831853f * u2);
}

// ---------------------------------------------------------------------------
// Host-side launch
// ---------------------------------------------------------------------------
extern "C" void kernel_launch(void* const* d_in, const int* in_sizes, int n_in,
                              void* d_out, int out_size, void* d_ws, size_t ws_size,
                              hipStream_t stream) {
    const float* H      = (const float*)d_in[0];   // [48, 200704]
    const float* bp     = (const float*)d_in[1];   // [48, 154]
    const float* w      = (const float*)d_in[2];   // [48, 3]
    const float* W_T1   = (const float*)d_in[3];   // [512, 154]
    const float* b_T1   = (const float*)d_in[4];
    const float* W_T2   = (const float*)d_in[5];   // [512, 3]
    const float* b_T2   = (const float*)d_in[6];
    const float* W_T3   = (const float*)d_in[7];   // [2048, 200704]
    const float* b_T3   = (const float*)d_in[8];
    const float* W_H3   = (const float*)d_in[9];   // [1024, 2048]
    const float* W_H4   = (const float*)d_in[10];
    const float* W_H5   = (const float*)d_in[11];
    const float* W_W1   = (const float*)d_in[12];  // [512, 512]
    const float* W_W2   = (const float*)d_in[13];
    const float* W_phi6 = (const float*)d_in[14];  // [5]
    const float* W_phi7 = (const float*)d_in[15];  // [2048, 1024]

    float* ws = (float*)d_ws;
    // workspace layout (float offsets)
    const size_t HF_ACC = 0;        // 48*2048
    const size_t HF     = 98304;    // 48*2048
    const size_t BPF    = 196608;   // 48*512
    const size_t WF     = 221184;   // 48*512
    const size_t P3     = 245760;   // 48*1024
    const size_t P4     = 294912;
    const size_t P5     = 344064;
    const size_t Q1     = 393216;   // 48*512
    const size_t Q2     = 417792;
    const size_t SMAT   = 442368;   // 80*9
    const size_t MUL    = 443088;   // 48*1024
    const size_t RACC   = 492240;   // 48*2048
    const size_t WS_FLOATS = 590544;

    // zero the accumulators (split-K partials land via atomics)
    hipMemsetAsync(d_ws, 0, WS_FLOATS * sizeof(float), stream);

    // 1) Hf_acc = H @ W_T3^T  (bandwidth-dominant; split-K 98 x 128 n-tiles)
    gemm48_nt_wmma<<<dim3(2048 / 16, 98), 96, 0, stream>>>(H, W_T3, ws + HF_ACC,
                                                           2048, 200704, 2048);
    // 2) Hf = relu(Hf_acc + b_T3)
    bias_relu48<<<(48 * 2048 + 255) / 256, 256, 0, stream>>>(ws + HF_ACC, b_T3,
                                                             ws + HF, 2048);
    // 3) tiny input projections
    proj_relu<<<(48 * 512 + 255) / 256, 256, 0, stream>>>(bp, W_T1, b_T1, ws + BPF, 154, 512);
    proj_relu<<<(48 * 512 + 255) / 256, 256, 0, stream>>>(w,  W_T2, b_T2, ws + WF,  3,   512);

    // 4) medium GEMMs (all via the same WMMA kernel)
    gemm48_nt_wmma<<<dim3(1024 / 16, 4), 96, 0, stream>>>(ws + HF, W_H3, ws + P3, 1024, 2048, 512);
    gemm48_nt_wmma<<<dim3(1024 / 16, 4), 96, 0, stream>>>(ws + HF, W_H4, ws + P4, 1024, 2048, 512);
    gemm48_nt_wmma<<<dim3(1024 / 16, 4), 96, 0, stream>>>(ws + HF, W_H5, ws + P5, 1024, 2048, 512);
    gemm48_nt_wmma<<<dim3(512 / 16, 1),  96, 0, stream>>>(ws + WF, W_W1, ws + Q1, 512,  512,  512);
    gemm48_nt_wmma<<<dim3(512 / 16, 1),  96, 0, stream>>>(ws + WF, W_W2, ws + Q2, 512,  512,  512);

    // 5) gram matrices + softmax -> stacked S[80][9]
    gram_softmax<<<80, 256, 0, stream>>>(ws + HF, ws + BPF, ws + WF,
                                         ws + Q1, ws + Q2, ws + P3, ws + P4, ws + SMAT);

    // 6) phi6 (batch-mixing reshape) and mul = phi6 @ P5
    phi6_mul<<<16, 256, 0, stream>>>(ws + SMAT, W_phi6, ws + P5, ws + MUL);

    // 7) Racc = mul @ W_phi7^T  [48 x 2048]
    gemm48_nt_wmma<<<dim3(2048 / 16, 4), 96, 0, stream>>>(ws + MUL, W_phi7, ws + RACC,
                                                          2048, 1024, 256);

    // 8) residual + transpose writeout (Z twice) + rand placeholder
    float* out = (float*)d_out;
    final_z<<<(48 * 2048 + 255) / 256, 256, 0, stream>>>(ws + HF, ws + RACC, out);
    fill_rand<<<1, 64, 0, stream>>>(out);
}